// PerturbationModel_47339129536881
// MI455X (gfx1250) — compile-verified
//
#include <hip/hip_runtime.h>
#include <math.h>

#define BATCH    65536
#define IN_DIM   512
#define LATENT   32
#define NUM_TRIL 528
#define BN_EPS   1e-5f
#define SLOPE    0.01f

#define WAVES_PER_BLOCK 2
#define ROWS_PER_WAVE   16
#define ROWS_PER_BLOCK  (WAVES_PER_BLOCK * ROWS_PER_WAVE)

#define H_STRIDE 33                        // h tile row stride (odd -> conflict-free)
#define H_TILE   (ROWS_PER_WAVE * H_STRIDE)        // 528 floats / wave

#define E_STRIDE 532                       // packed-tri row stride (8*532 == 32 mod 64:
                                           //  the two lane-halves of a scatter store
                                           //  hit disjoint bank sets)
#define E_TILE   (ROWS_PER_WAVE * E_STRIDE)        // 8512 floats / wave

#define SMEM_FLOATS (WAVES_PER_BLOCK * (H_TILE + E_TILE))   // ~72 KB / block

typedef float v2f __attribute__((ext_vector_type(2)));
typedef float v8f __attribute__((ext_vector_type(8)));

__device__ __forceinline__ v8f wmma_f32_4(v2f a, v2f b, v8f c) {
  // D = A(16x4 f32) * B(4x16 f32) + C(16x16 f32)
  return __builtin_amdgcn_wmma_f32_16x16x4_f32(
      /*neg_a=*/false, a, /*neg_b=*/false, b,
      /*c_mod=*/(short)0, c, /*reuse_a=*/false, /*reuse_b=*/false);
}

__device__ __forceinline__ float lrelu(float v) {
  return v >= 0.0f ? v : SLOPE * v;
}

__global__ __launch_bounds__(WAVES_PER_BLOCK * 32)
void perturb_fused_kernel(const float* __restrict__ x,
                          const float* __restrict__ W_enc,
                          const float* __restrict__ b_enc,
                          const float* __restrict__ bn_gamma,
                          const float* __restrict__ bn_beta,
                          const float* __restrict__ bn_mean,
                          const float* __restrict__ bn_var,
                          const float* __restrict__ W_mu,
                          const float* __restrict__ b_mu,
                          const float* __restrict__ W_ch,
                          const float* __restrict__ b_ch,
                          float* __restrict__ h_out,
                          float* __restrict__ mu_out,
                          float* __restrict__ cov_out)
{
  extern __shared__ float smem[];
  const int lane  = threadIdx.x & 31;
  const int wave  = threadIdx.x >> 5;
  const int rbase = (blockIdx.x * WAVES_PER_BLOCK + wave) * ROWS_PER_WAVE;

  float* hS = smem + wave * H_TILE;
  float* eS = smem + WAVES_PER_BLOCK * H_TILE + wave * E_TILE;  // packed tri elts, 16 rows

  const int mrow  = lane & 15;          // A row / B col within a 16-tile
  const int khalf = (lane >> 4) * 2;    // K sub-offset: lanes 0-15 -> {0,1}, 16-31 -> {2,3}
  const int rhalf = (lane >> 4) * 8;    // C/D: lanes 16-31 hold M = 8..15

  // ---------------- Stage 1: h = leaky(BN(x @ W_enc + b_enc)) ----------------
  v8f acc0 = {0.f,0.f,0.f,0.f,0.f,0.f,0.f,0.f};
  v8f acc1 = acc0;
  const float* xrow = x + (size_t)(rbase + mrow) * IN_DIM;

  #pragma unroll 2
  for (int k0 = 0; k0 < IN_DIM; k0 += 4) {
    const float2 av = *reinterpret_cast<const float2*>(xrow + k0 + khalf);
    v2f a; a.x = av.x; a.y = av.y;
    v2f b0, b1;
    b0.x = W_enc[(k0 + khalf    ) * LATENT + mrow];
    b0.y = W_enc[(k0 + khalf + 1) * LATENT + mrow];
    b1.x = W_enc[(k0 + khalf    ) * LATENT + mrow + 16];
    b1.y = W_enc[(k0 + khalf + 1) * LATENT + mrow + 16];
    acc0 = wmma_f32_4(a, b0, acc0);
    acc1 = wmma_f32_4(a, b1, acc1);
  }

  // BN + leaky epilogue; write h to HBM and LDS
  #pragma unroll
  for (int t = 0; t < 2; ++t) {
    const int n  = t * 16 + mrow;
    const float be = b_enc[n];
    const float sc = bn_gamma[n] * rsqrtf(bn_var[n] + BN_EPS);
    const float sh = bn_beta[n] - bn_mean[n] * sc;
    v8f acc = t == 0 ? acc0 : acc1;
    #pragma unroll
    for (int j = 0; j < 8; ++j) {
      const int rit = j + rhalf;
      float v = lrelu((acc[j] + be) * sc + sh);
      h_out[(size_t)(rbase + rit) * LATENT + n] = v;
      hS[rit * H_STRIDE + n] = v;
    }
  }
  __syncthreads();

  // Preload the 8 A-fragments of this wave's 16x32 h-tile (reused for mus & elts)
  v2f ah[8];
  #pragma unroll
  for (int j = 0; j < 8; ++j) {
    const int k = j * 4 + khalf;
    ah[j].x = hS[mrow * H_STRIDE + k];
    ah[j].y = hS[mrow * H_STRIDE + k + 1];
  }

  // ---------------- Stage 2: mus = leaky(h @ W_mu + b_mu) ----------------
  {
    v8f m0 = {0.f,0.f,0.f,0.f,0.f,0.f,0.f,0.f};
    v8f m1 = m0;
    #pragma unroll
    for (int j = 0; j < 8; ++j) {
      const int k = j * 4 + khalf;
      v2f b0, b1;
      b0.x = W_mu[(k    ) * LATENT + mrow];
      b0.y = W_mu[(k + 1) * LATENT + mrow];
      b1.x = W_mu[(k    ) * LATENT + mrow + 16];
      b1.y = W_mu[(k + 1) * LATENT + mrow + 16];
      m0 = wmma_f32_4(ah[j], b0, m0);
      m1 = wmma_f32_4(ah[j], b1, m1);
    }
    #pragma unroll
    for (int t = 0; t < 2; ++t) {
      const int n = t * 16 + mrow;
      const float bm = b_mu[n];
      v8f acc = t == 0 ? m0 : m1;
      #pragma unroll
      for (int j = 0; j < 8; ++j) {
        const int rit = j + rhalf;
        mu_out[(size_t)(rbase + rit) * LATENT + n] = lrelu(acc[j] + bm);
      }
    }
  }

  // ------- Stage 3: elts = act(h @ W_ch + b_ch) into packed-tri LDS rows -------
  #pragma unroll 1
  for (int t = 0; t < NUM_TRIL / 16; ++t) {    // 33 tiles of 16 columns
    const int c = t * 16 + mrow;               // packed-tri column handled by this lane
    v8f acc = {0.f,0.f,0.f,0.f,0.f,0.f,0.f,0.f};
    #pragma unroll
    for (int j = 0; j < 8; ++j) {
      const int k = j * 4 + khalf;
      v2f b;
      b.x = W_ch[(k    ) * NUM_TRIL + c];
      b.y = W_ch[(k + 1) * NUM_TRIL + c];
      acc = wmma_f32_4(ah[j], b, acc);
    }
    // is column c the diagonal slot of its triangular row?  T(i) <= c < T(i+1)
    int i = (int)((sqrtf(8.0f * (float)c + 1.0f) - 1.0f) * 0.5f);
    while (((i + 1) * (i + 2)) / 2 <= c) ++i;
    while ((i * (i + 1)) / 2 > c) --i;
    const bool isdiag = (c - (i * (i + 1)) / 2 == i);
    const float bc = b_ch[c];
    #pragma unroll
    for (int j = 0; j < 8; ++j) {
      const int rit = j + rhalf;
      float v = acc[j] + bc;
      if (isdiag) {
        float sp = (v > 20.0f) ? v : log1pf(expf(v));  // softplus
        sp = fminf(fmaxf(sp, 0.001f), 100.0f);
        v = sp + 0.01f;
      } else {
        v = lrelu(v);
      }
      eS[rit * E_STRIDE + c] = v;
    }
  }
  __syncthreads();

  // ---------------- Stage 4: cov = L @ L^T + 0.01*I via WMMA ----------------
  // Fragments come straight from the packed triangular row: element (row, col=k+v)
  // is packed[T(row)+k+v] when col <= row, else 0 (predicated with cndmask; the
  // unconditional loads stay inside the 532-float padded row).
  // B = A^T and the ISA B-layout is the transpose of the A-layout, so the A-frag
  // of an L row-tile doubles as the B-frag of the matching L^T col-tile.
  const int o0 = (mrow * (mrow + 1)) >> 1;                    // T(mrow),    rows 0..15
  const int o1 = ((mrow + 16) * (mrow + 17)) >> 1;            // T(mrow+16), rows 16..31
  #pragma unroll 1
  for (int r = 0; r < ROWS_PER_WAVE; ++r) {
    const float* Lp = eS + r * E_STRIDE;
    float* crow = cov_out + (size_t)(rbase + r) * (LATENT * LATENT);
    v8f c00 = {0.f,0.f,0.f,0.f,0.f,0.f,0.f,0.f};
    v8f c01 = c00, c10 = c00, c11 = c00;
    #pragma unroll
    for (int ks = 0; ks < 8; ++ks) {
      const int k = ks * 4 + khalf;
      v2f f0, f1;
      f0.x = (k     <= mrow) ? Lp[o0 + k    ] : 0.0f;
      f0.y = (k + 1 <= mrow) ? Lp[o0 + k + 1] : 0.0f;
      f1.x = (k     <= mrow + 16) ? Lp[o1 + k    ] : 0.0f;
      f1.y = (k + 1 <= mrow + 16) ? Lp[o1 + k + 1] : 0.0f;
      c00 = wmma_f32_4(f0, f0, c00);
      c01 = wmma_f32_4(f0, f1, c01);
      c10 = wmma_f32_4(f1, f0, c10);
      c11 = wmma_f32_4(f1, f1, c11);
    }
    #pragma unroll
    for (int j = 0; j < 8; ++j) {
      const int m = j + rhalf;
      const float eye = (m == mrow) ? 0.01f : 0.0f;
      crow[(m     ) * LATENT + mrow     ] = c00[j] + eye;
      crow[(m     ) * LATENT + mrow + 16] = c01[j];
      crow[(m + 16) * LATENT + mrow     ] = c10[j];
      crow[(m + 16) * LATENT + mrow + 16] = c11[j] + eye;
    }
  }
}

extern "C" void kernel_launch(void* const* d_in, const int* in_sizes, int n_in,
                              void* d_out, int out_size, void* d_ws, size_t ws_size,
                              hipStream_t stream) {
  const float* x        = (const float*)d_in[0];
  const float* W_enc    = (const float*)d_in[1];
  const float* b_enc    = (const float*)d_in[2];
  const float* bn_gamma = (const float*)d_in[3];
  const float* bn_beta  = (const float*)d_in[4];
  const float* bn_mean  = (const float*)d_in[5];
  const float* bn_var   = (const float*)d_in[6];
  const float* W_mu     = (const float*)d_in[7];
  const float* b_mu     = (const float*)d_in[8];
  const float* W_ch     = (const float*)d_in[9];
  const float* b_ch     = (const float*)d_in[10];

  float* out     = (float*)d_out;
  float* h_out   = out;
  float* mu_out  = out + (size_t)BATCH * LATENT;
  float* cov_out = out + (size_t)2 * BATCH * LATENT;

  const dim3 grid(BATCH / ROWS_PER_BLOCK);
  const dim3 block(WAVES_PER_BLOCK * 32);
  const size_t shmem = (size_t)SMEM_FLOATS * sizeof(float);

  hipLaunchKernelGGL(perturb_fused_kernel, grid, block, shmem, stream,
                     x, W_enc, b_enc, bn_gamma, bn_beta, bn_mean, bn_var,
                     W_mu, b_mu, W_ch, b_ch, h_out, mu_out, cov_out);
}